// MultiHeadEdgeGATLayer_24575802867852
// MI455X (gfx1250) — compile-verified
//
#include <hip/hip_runtime.h>
#include <hip/hip_bf16.h>
#include <math.h>

typedef __attribute__((ext_vector_type(2))) float v2f;
typedef __attribute__((ext_vector_type(8))) float v8f;

#define N_NODES 50000
#define N_EDGES 800000
#define F_NODE 128
#define F_EDGE 32
#define HEADS 8
#define F_OUT 16
#define HID 128
#define SLOPE 0.2f
#define LN_EPS 1e-5f
#define WPB 4            // waves per block in GEMM kernels
#define LSTRIDE 132      // LDS row stride (floats), breaks 16-way bank conflicts

// ---------------------------------------------------------------------------
// float atomic max via CAS (init value is -3e38, so plain CAS loop is fine)
// ---------------------------------------------------------------------------
__device__ __forceinline__ void atomicMaxF(float* addr, float v) {
    unsigned int* ua = (unsigned int*)addr;
    unsigned int old = *ua;
    while (__uint_as_float(old) < v) {
        unsigned int assumed = old;
        old = atomicCAS(ua, assumed, __float_as_uint(v));
        if (old == assumed) break;
    }
}

// ---------------------------------------------------------------------------
// Kernel 1: hn = X @ W^T + b  (50000x128 @ 128x128), fp32 WMMA 16x16x4.
// Epilogue (per 16-row tile in LDS): store hn, and per-head attention
// scalars s1 = <hn, a_w[0:16]>, s2 = <hn, a_w[16:32]>.
// ---------------------------------------------------------------------------
__global__ __launch_bounds__(128) void gat_node_gemm(
    const float* __restrict__ X, const float* __restrict__ W,
    const float* __restrict__ bias, const float* __restrict__ a_w,
    float* __restrict__ hn, float* __restrict__ s1, float* __restrict__ s2)
{
    __shared__ float lds[WPB * 16 * LSTRIDE];
    const int lane  = threadIdx.x & 31;
    const int wave  = threadIdx.x >> 5;
    const int gwave = blockIdx.x * WPB + wave;
    if (gwave >= N_NODES / 16) return;
    const int row0 = gwave * 16;
    const int m    = lane & 15;            // M (for A) / N (for B) within tile
    const int kk   = (lane >> 4) << 1;     // K sub-offset: 0 or 2

    v8f acc[8] = {};
    const float* arow = X + (size_t)(row0 + m) * F_NODE + kk;
    for (int k0 = 0; k0 < F_NODE; k0 += 4) {
        v2f a; a.x = arow[k0]; a.y = arow[k0 + 1];
#pragma unroll
        for (int t = 0; t < 8; ++t) {
            const float* brow = W + (size_t)(t * 16 + m) * F_NODE + k0 + kk;
            v2f b; b.x = brow[0]; b.y = brow[1];
            acc[t] = __builtin_amdgcn_wmma_f32_16x16x4_f32(
                false, a, false, b, (short)0, acc[t], false, false);
        }
    }

    // C layout: vgpr r -> row r (lanes 0-15) / row r+8 (lanes 16-31), col = t*16 + (lane&15)
    float* my = lds + wave * (16 * LSTRIDE);
    const int rbase = (lane >> 4) << 3;
#pragma unroll
    for (int t = 0; t < 8; ++t)
#pragma unroll
        for (int r = 0; r < 8; ++r)
            my[(rbase + r) * LSTRIDE + t * 16 + m] = acc[t][r];

    if (lane < 16) {
        const int node = row0 + lane;
        const float* rowp = my + lane * LSTRIDE;
        float* hrow = hn + (size_t)node * HID;
#pragma unroll
        for (int h = 0; h < HEADS; ++h) {
            float d1 = 0.f, d2 = 0.f;
#pragma unroll
            for (int f = 0; f < F_OUT; ++f) {
                float v = rowp[h * 16 + f] + bias[h * 16 + f];
                hrow[h * 16 + f] = v;
                d1 += v * a_w[f];
                d2 += v * a_w[F_OUT + f];
            }
            s1[node * HEADS + h] = d1;
            s2[node * HEADS + h] = d2;
        }
    }
}

// ---------------------------------------------------------------------------
// Kernel 2: edge_h = E @ We^T + be (800000x32 @ 128x32), fp32 WMMA.
// edge_h is never written to global: epilogue emits elog = <edge_h, a_w[32:48]>
// per head, and LayerNorm(edge_h) straight into the edge output.
// ---------------------------------------------------------------------------
__global__ __launch_bounds__(128) void gat_edge_gemm(
    const float* __restrict__ E, const float* __restrict__ We,
    const float* __restrict__ be, const float* __restrict__ a_w,
    const float* __restrict__ gamma, const float* __restrict__ beta,
    float* __restrict__ elog, float* __restrict__ edge_out)
{
    __shared__ float lds[WPB * 16 * LSTRIDE];
    const int lane  = threadIdx.x & 31;
    const int wave  = threadIdx.x >> 5;
    const int gwave = blockIdx.x * WPB + wave;
    if (gwave >= N_EDGES / 16) return;
    const int row0 = gwave * 16;
    const int m    = lane & 15;
    const int kk   = (lane >> 4) << 1;

    v8f acc[8] = {};
    const float* arow = E + (size_t)(row0 + m) * F_EDGE + kk;
#pragma unroll
    for (int k0 = 0; k0 < F_EDGE; k0 += 4) {
        v2f a; a.x = arow[k0]; a.y = arow[k0 + 1];
#pragma unroll
        for (int t = 0; t < 8; ++t) {
            const float* brow = We + (size_t)(t * 16 + m) * F_EDGE + k0 + kk;
            v2f b; b.x = brow[0]; b.y = brow[1];
            acc[t] = __builtin_amdgcn_wmma_f32_16x16x4_f32(
                false, a, false, b, (short)0, acc[t], false, false);
        }
    }

    float* my = lds + wave * (16 * LSTRIDE);
    const int rbase = (lane >> 4) << 3;
#pragma unroll
    for (int t = 0; t < 8; ++t)
#pragma unroll
        for (int r = 0; r < 8; ++r)
            my[(rbase + r) * LSTRIDE + t * 16 + m] = acc[t][r];

    if (lane < 16) {
        const int e = row0 + lane;
        const float* rowp = my + lane * LSTRIDE;
        float sum = 0.f, sumsq = 0.f;
#pragma unroll
        for (int h = 0; h < HEADS; ++h) {
            float d3 = 0.f;
#pragma unroll
            for (int f = 0; f < F_OUT; ++f) {
                float v = rowp[h * 16 + f] + be[h * 16 + f];
                sum += v; sumsq += v * v;
                d3 += v * a_w[2 * F_OUT + f];
            }
            elog[e * HEADS + h] = d3;
        }
        float mean = sum * (1.f / HID);
        float var  = sumsq * (1.f / HID) - mean * mean;
        float inv  = rsqrtf(var + LN_EPS);
        float* orow = edge_out + (size_t)e * HID;
#pragma unroll
        for (int c = 0; c < HID; ++c) {
            float v = rowp[c] + be[c];
            orow[c] = (v - mean) * inv * gamma[c] + beta[c];
        }
    }
}

// ---------------------------------------------------------------------------
// Kernel 3: reset accumulators (must run every launch; ws is not re-poisoned)
// ---------------------------------------------------------------------------
__global__ void gat_init(float* out_acc, float* denom, float* emax, float* deg)
{
    int i = blockIdx.x * blockDim.x + threadIdx.x;
    if (i < N_NODES * HID)   out_acc[i] = 0.f;
    if (i < N_NODES * HEADS) { denom[i] = 0.f; emax[i] = -3.0e38f; }
    if (i < N_NODES)         deg[i] = 0.f;
}

// ---------------------------------------------------------------------------
// Kernel 4: per-(edge,head) logits, leaky-relu, segment max, degree count
// ---------------------------------------------------------------------------
__global__ void gat_edge_logits(
    const int* __restrict__ src, const int* __restrict__ tgt,
    const float* __restrict__ s1, const float* __restrict__ s2,
    const float* __restrict__ elog, const float* __restrict__ a_b,
    float* __restrict__ ebuf, float* __restrict__ emax, float* __restrict__ deg)
{
    int i = blockIdx.x * blockDim.x + threadIdx.x;
    if (i >= N_EDGES * HEADS) return;
    int e = i >> 3, h = i & 7;
    int t = tgt[e], s = src[e];
    float x = s1[t * HEADS + h] + s2[s * HEADS + h] + elog[i] + a_b[0];
    x = x > 0.f ? x : SLOPE * x;
    ebuf[i] = x;
    atomicMaxF(emax + t * HEADS + h, x);
    if (h == 0) atomicAdd(deg + t, 1.0f);
}

// ---------------------------------------------------------------------------
// Kernel 5: exp(e - max) and segment sum (softmax denominator)
// ---------------------------------------------------------------------------
__global__ void gat_edge_exp(
    const int* __restrict__ tgt, const float* __restrict__ emax,
    float* __restrict__ ebuf, float* __restrict__ denom)
{
    int i = blockIdx.x * blockDim.x + threadIdx.x;
    if (i >= N_EDGES * HEADS) return;
    int e = i >> 3, h = i & 7;
    int t = tgt[e];
    float ex = expf(ebuf[i] - emax[t * HEADS + h]);
    ebuf[i] = ex;
    atomicAdd(denom + t * HEADS + h, ex);
}

// ---------------------------------------------------------------------------
// Kernel 6: one wave per edge: out_acc[tgt] += alpha * hn[src]
// (the per-edge +h_i term is folded into deg(i)*hn[i] in the finalize kernel)
// ---------------------------------------------------------------------------
__global__ __launch_bounds__(256) void gat_edge_aggregate(
    const int* __restrict__ src, const int* __restrict__ tgt,
    const float* __restrict__ hn, const float* __restrict__ ebuf,
    const float* __restrict__ denom, float* __restrict__ out_acc)
{
    int gw   = (blockIdx.x * blockDim.x + threadIdx.x) >> 5;
    int lane = threadIdx.x & 31;
    if (gw >= N_EDGES) return;
    int s = src[gw], t = tgt[gw];
    int h = lane >> 2;                         // 4 consecutive cols share a head
    float alpha = ebuf[gw * HEADS + h] / denom[t * HEADS + h];
    const float4 v = ((const float4*)(hn + (size_t)s * HID))[lane];
    float* o = out_acc + (size_t)t * HID + lane * 4;
    atomicAdd(o + 0, alpha * v.x);
    atomicAdd(o + 1, alpha * v.y);
    atomicAdd(o + 2, alpha * v.z);
    atomicAdd(o + 3, alpha * v.w);
}

// ---------------------------------------------------------------------------
// Kernel 7: one wave per node: val = out_acc + deg*hn; LayerNorm; ELU
// ---------------------------------------------------------------------------
__global__ __launch_bounds__(256) void gat_node_finalize(
    const float* __restrict__ out_acc, const float* __restrict__ hn,
    const float* __restrict__ deg, const float* __restrict__ gamma,
    const float* __restrict__ beta, float* __restrict__ out)
{
    int gw   = (blockIdx.x * blockDim.x + threadIdx.x) >> 5;
    int lane = threadIdx.x & 31;
    if (gw >= N_NODES) return;
    float d = deg[gw];
    float4 a = ((const float4*)(out_acc + (size_t)gw * HID))[lane];
    float4 h = ((const float4*)(hn + (size_t)gw * HID))[lane];
    float4 v;
    v.x = a.x + d * h.x; v.y = a.y + d * h.y;
    v.z = a.z + d * h.z; v.w = a.w + d * h.w;
    float s  = v.x + v.y + v.z + v.w;
    float sq = v.x * v.x + v.y * v.y + v.z * v.z + v.w * v.w;
#pragma unroll
    for (int off = 16; off > 0; off >>= 1) {
        s  += __shfl_xor(s, off, 32);
        sq += __shfl_xor(sq, off, 32);
    }
    float mean = s * (1.f / HID);
    float var  = sq * (1.f / HID) - mean * mean;
    float inv  = rsqrtf(var + LN_EPS);
    int c = lane * 4;
    float4 r;
    r.x = (v.x - mean) * inv * gamma[c + 0] + beta[c + 0];
    r.y = (v.y - mean) * inv * gamma[c + 1] + beta[c + 1];
    r.z = (v.z - mean) * inv * gamma[c + 2] + beta[c + 2];
    r.w = (v.w - mean) * inv * gamma[c + 3] + beta[c + 3];
    r.x = r.x > 0.f ? r.x : expf(r.x) - 1.f;
    r.y = r.y > 0.f ? r.y : expf(r.y) - 1.f;
    r.z = r.z > 0.f ? r.z : expf(r.z) - 1.f;
    r.w = r.w > 0.f ? r.w : expf(r.w) - 1.f;
    ((float4*)(out + (size_t)gw * HID))[lane] = r;
}

// ---------------------------------------------------------------------------
extern "C" void kernel_launch(void* const* d_in, const int* in_sizes, int n_in,
                              void* d_out, int out_size, void* d_ws, size_t ws_size,
                              hipStream_t stream) {
    const float* node_features = (const float*)d_in[0];
    const float* edge_features = (const float*)d_in[1];
    const int*   edge_index    = (const int*)d_in[2];
    const float* W     = (const float*)d_in[3];
    const float* b     = (const float*)d_in[4];
    const float* We    = (const float*)d_in[5];
    const float* be    = (const float*)d_in[6];
    const float* a_w   = (const float*)d_in[7];
    const float* a_b   = (const float*)d_in[8];
    const float* gamma = (const float*)d_in[9];
    const float* beta  = (const float*)d_in[10];

    const int* src = edge_index;             // edge_index[0]
    const int* tgt = edge_index + N_EDGES;   // edge_index[1]

    float* out_nodes = (float*)d_out;
    float* out_edges = out_nodes + (size_t)N_NODES * HID;

    // workspace carve-up (floats)
    float* ws      = (float*)d_ws;
    float* hn      = ws;                                 // N_NODES*HID   = 6.4M
    float* s1      = hn      + (size_t)N_NODES * HID;    // N_NODES*HEADS = 0.4M
    float* s2      = s1      + (size_t)N_NODES * HEADS;
    float* elog    = s2      + (size_t)N_NODES * HEADS;  // N_EDGES*HEADS = 6.4M
    float* ebuf    = elog    + (size_t)N_EDGES * HEADS;  // N_EDGES*HEADS = 6.4M
    float* emax    = ebuf    + (size_t)N_EDGES * HEADS;  // N_NODES*HEADS
    float* denom   = emax    + (size_t)N_NODES * HEADS;  // N_NODES*HEADS
    float* deg     = denom   + (size_t)N_NODES * HEADS;  // N_NODES
    float* out_acc = deg     + (size_t)N_NODES;          // N_NODES*HID

    // 1) node GEMM + attention scalars (WMMA f32)
    {
        int tiles = N_NODES / 16;                        // 3125
        int blocks = (tiles + WPB - 1) / WPB;
        gat_node_gemm<<<blocks, 128, 0, stream>>>(node_features, W, b, a_w, hn, s1, s2);
    }
    // 2) edge GEMM + elog + LayerNorm'd edge output (WMMA f32)
    {
        int tiles = N_EDGES / 16;                        // 50000
        int blocks = (tiles + WPB - 1) / WPB;
        gat_edge_gemm<<<blocks, 128, 0, stream>>>(edge_features, We, be, a_w,
                                                  gamma, beta, elog, out_edges);
    }
    // 3) reset accumulators
    {
        int n = N_NODES * HID;
        gat_init<<<(n + 255) / 256, 256, 0, stream>>>(out_acc, denom, emax, deg);
    }
    // 4) logits + segment max + degree
    {
        int n = N_EDGES * HEADS;
        gat_edge_logits<<<(n + 255) / 256, 256, 0, stream>>>(src, tgt, s1, s2, elog,
                                                             a_b, ebuf, emax, deg);
    }
    // 5) exp + segment sum
    {
        int n = N_EDGES * HEADS;
        gat_edge_exp<<<(n + 255) / 256, 256, 0, stream>>>(tgt, emax, ebuf, denom);
    }
    // 6) weighted scatter aggregation (wave per edge)
    {
        long long thr = (long long)N_EDGES * 32;
        gat_edge_aggregate<<<(int)((thr + 255) / 256), 256, 0, stream>>>(
            src, tgt, hn, ebuf, denom, out_acc);
    }
    // 7) finalize: +deg*hn, LayerNorm, ELU (wave per node)
    {
        long long thr = (long long)N_NODES * 32;
        gat_node_finalize<<<(int)((thr + 255) / 256), 256, 0, stream>>>(
            out_acc, hn, deg, gamma, beta, out_nodes);
    }
}